// AdExNeuron_39032662786072
// MI455X (gfx1250) — compile-verified
//
#include <hip/hip_runtime.h>

// ---------------- Problem constants (from reference) ----------------
#define T_STEPS   2048
#define N_NEURONS 32768

// AdEx constants. With A=0, B=0, w0=0 the adaptation variable w is
// identically zero for all time, so it is eliminated.
// TAU_M = C/G_L = 20, DT = 0.1 -> DT/TAU_M = 0.005
#define K_EL       (-70.0f)   // E_L
#define K_VT       (0.6f)     // V_T
#define K_INV_DT2  (0.5f)     // 1/DELTA_T
#define K_DELTA_T  (2.0f)     // DELTA_T
#define K_STEP     (0.005f)   // DT / TAU_M
#define K_VSPIKE   (30.0f)
#define K_VRESET   (-65.0f)

// ---------------- Tiling / async pipeline ----------------
#define BLK        256              // threads per block (8 wave32s)
#define DEPTH      62               // async prefetch depth (ASYNCcnt is 6-bit, <=63)
#define SLOTS      (DEPTH + 1)      // +1 slack: prefetch never hits the slot being read
#define SLOT_BYTES (BLK * 4)        // 1 KB per staged timestep per block
#define WRAP_BYTES (SLOTS * SLOT_BYTES)

// LDS byte offset of a pointer into a __shared__ array (addrspace(3) is 32-bit).
__device__ __forceinline__ unsigned lds_addr_u32(const void* p) {
  return (unsigned)(unsigned long long)
      (__attribute__((address_space(3))) const char*)p;
}

// gfx1250 async copy, GVS form: addr = SGPR64 base + per-lane 32-bit voffset.
// 4 bytes per lane -> LDS, tracked by ASYNCcnt. Base is uniform -> SALU walks it.
__device__ __forceinline__ void async_load_f32_to_lds(unsigned lds_byte_addr,
                                                      unsigned voff_bytes,
                                                      const float* sbase) {
  asm volatile("global_load_async_to_lds_b32 %0, %1, %2"
               :
               : "v"(lds_byte_addr),
                 "v"(voff_bytes),
                 "s"((unsigned long long)(uintptr_t)sbase)
               : "memory");
}

#define WAIT_ASYNCCNT(n) asm volatile("s_wait_asynccnt " #n ::: "memory")

// One AdEx Euler step (w == 0 eliminated). Returns spike (1.0f / 0.0f).
__device__ __forceinline__ float adex_step(float& V, float inp) {
  // exp_term = DELTA_T * exp((V - V_T)/DELTA_T)
  const float e   = __expf((V - K_VT) * K_INV_DT2);   // v_exp_f32 path
  const float acc = (K_EL - V) + K_DELTA_T * e + inp; // -(V-E_L)+exp_term+R*I
  V = fmaf(K_STEP, acc, V);                           // V += DT*dV
  const bool  fired = (V >= K_VSPIKE);
  const float spike = fired ? 1.0f : 0.0f;
  V = fired ? K_VRESET : V;
  return spike;
}

__global__ __launch_bounds__(BLK)
void AdEx_async_scan_kernel(const float* __restrict__ I,
                            float* __restrict__ S) {
  __shared__ float buf[SLOTS * BLK];   // 63 KB circular staging buffer

  const int tid       = threadIdx.x;
  const int gid       = blockIdx.x * BLK + tid;   // neuron id (grid sized exactly)
  const unsigned voff = (unsigned)gid * 4u;       // per-lane byte offset (fixed VGPR)
  const unsigned lbase = lds_addr_u32(&buf[tid]);

  const float* base = I;               // uniform row base: I + t*N (SGPR walk)

  // Prologue: fill the pipeline with DEPTH prefetches (slots 0..DEPTH-1).
#pragma unroll 4
  for (int k = 0; k < DEPTH; ++k) {
    async_load_f32_to_lds(lbase + (unsigned)(k * SLOT_BYTES), voff, base);
    base += N_NEURONS;
  }

  float V = K_EL;            // V0 = E_L
  int      cs   = 0;         // consume slot index        = t         mod SLOTS
  unsigned poff = (unsigned)(DEPTH * SLOT_BYTES);  // prefetch slot byte offset

  // Main loop: steady state, DEPTH async loads always in flight.
  for (int t = 0; t < T_STEPS - DEPTH; ++t) {
    // In-order completion: <= DEPTH-1 outstanding => prefetch(t) has landed.
    WAIT_ASYNCCNT(61);
    const float x     = buf[cs * BLK + tid];          // ds_load_b32
    const float spike = adex_step(V, x);

    float* row = S + (size_t)t * N_NEURONS;           // uniform -> saddr store form
    __builtin_nontemporal_store(spike, row + gid);    // streaming store (TH=NT)

    // Refill: prefetch timestep t+DEPTH (slot consumed at t-1; safe by SLOTS=D+1).
    async_load_f32_to_lds(lbase + poff, voff, base);
    base += N_NEURONS;

    cs   = (cs + 1 == SLOTS) ? 0 : cs + 1;
    poff = (poff + SLOT_BYTES == WRAP_BYTES) ? 0u : poff + SLOT_BYTES;
  }

  // Tail: drain the remaining DEPTH staged timesteps.
  WAIT_ASYNCCNT(0);
  for (int t = T_STEPS - DEPTH; t < T_STEPS; ++t) {
    const float x     = buf[cs * BLK + tid];
    const float spike = adex_step(V, x);
    float* row = S + (size_t)t * N_NEURONS;
    __builtin_nontemporal_store(spike, row + gid);
    cs = (cs + 1 == SLOTS) ? 0 : cs + 1;
  }
}

extern "C" void kernel_launch(void* const* d_in, const int* in_sizes, int n_in,
                              void* d_out, int out_size, void* d_ws, size_t ws_size,
                              hipStream_t stream) {
  (void)in_sizes; (void)n_in; (void)out_size; (void)d_ws; (void)ws_size;
  const float* I = (const float*)d_in[0];      // [T_STEPS, N_NEURONS] f32
  float*       S = (float*)d_out;              // [T_STEPS, N_NEURONS] f32 spikes
  AdEx_async_scan_kernel<<<dim3(N_NEURONS / BLK), dim3(BLK), 0, stream>>>(I, S);
}